// NeuroPlasticityMechanism_52725018526315
// MI455X (gfx1250) — compile-verified
//
#include <hip/hip_runtime.h>

typedef __attribute__((ext_vector_type(2))) float v2f;
typedef __attribute__((ext_vector_type(4))) float v4f;
typedef __attribute__((ext_vector_type(8))) float v8f;

#define TRACE_DECAY 0.9f
#define MOM_FACTOR  0.95f
#define MOD_STR     0.1f
#define LR_SCALE    0.1f

// Fused: [new_trace, new_mom, mod] with mod = r*0.1*new_mom + 0.1*(U@V)
// Block = 256 threads (8 waves). Wave w computes 16x16 tile at
// (blockIdx.y*16, blockIdx.x*128 + w*16). K = R = 64 via 16x WMMA_F32_16X16X4_F32.
// N is a template constant so all V-row / elementwise row offsets fold into
// 24-bit instruction offsets (no runtime 64-bit address multiplies).
template <int NN>
__global__ __launch_bounds__(256)
void plastic_mat_wmma(const float* __restrict__ grad,
                      const float* __restrict__ trace,
                      const float* __restrict__ mom,
                      const float* __restrict__ U,   // [M, 64]
                      const float* __restrict__ V,   // [64, NN]
                      const float* __restrict__ reward,
                      float* __restrict__ t_out,
                      float* __restrict__ m_out,
                      float* __restrict__ mod_out)
{
    const int R    = 64;
    const int wave = (int)threadIdx.x >> 5;
    const int lane = (int)threadIdx.x & 31;
    const int lo   = lane & 15;   // row (A) / col (B,C,D) within tile
    const int hi   = lane >> 4;   // half-wave select

    const int tileM = (int)blockIdx.y << 4;
    const int n0    = ((int)blockIdx.x << 7) + (wave << 4);

    v8f acc = {0.f, 0.f, 0.f, 0.f, 0.f, 0.f, 0.f, 0.f};

    // A (16x4 fp32): VGPR0 = K+0 (lanes 0-15) / K+2 (lanes 16-31), VGPR1 = K+1 / K+3
    //   -> per lane a contiguous float2 from a U row (8B aligned), imm offsets.
    const float* Urow = U + (unsigned)(tileM + lo) * R + (hi << 1);
    // B (4x16 fp32): VGPR0 = row K+0/K+2, VGPR1 = row K+1/K+3, N striped across lanes.
    // Single base; step/row deltas are compile-time (kb*NN, +NN) -> imm offsets.
    const float* Vbase = V + (unsigned)(hi << 1) * NN + (unsigned)(n0 + lo);

    #pragma unroll
    for (int k = 0; k < 16; ++k) {
        const int kb = k << 2;
        const v2f a = *(const v2f*)(Urow + kb);
        v2f b;
        b.x = Vbase[kb * NN];
        b.y = Vbase[kb * NN + NN];
        acc = __builtin_amdgcn_wmma_f32_16x16x4_f32(
            /*neg_a=*/false, a, /*neg_b=*/false, b,
            /*c_mod=*/(short)0, acc, /*reuse_a=*/false, /*reuse_b=*/false);
    }

    const float r = reward[0] * MOD_STR;   // uniform scalar load

    // C/D layout: VGPR j -> M = j + 8*hi, N = lo. Fuse elementwise update.
    // Streamed operands: non-temporal (read/written exactly once).
    const unsigned base = (unsigned)(tileM + (hi << 3)) * NN + (unsigned)(n0 + lo);
    #pragma unroll
    for (int j = 0; j < 8; ++j) {
        const unsigned idx = base + (unsigned)(j * NN);
        const float g  = __builtin_nontemporal_load(grad  + idx);
        const float t  = __builtin_nontemporal_load(trace + idx);
        const float m  = __builtin_nontemporal_load(mom   + idx);
        const float nt = TRACE_DECAY * t + (1.0f - TRACE_DECAY) * g;
        const float nm = MOM_FACTOR  * m + (1.0f - MOM_FACTOR)  * nt;
        __builtin_nontemporal_store(nt, t_out + idx);
        __builtin_nontemporal_store(nm, m_out + idx);
        __builtin_nontemporal_store(r * nm + LR_SCALE * acc[j], mod_out + idx);
    }
}

// 1-D bias path: no low-rank term, vectorized with native clang float4 so the
// nontemporal builtins accept the pointers (B128 NT loads/stores).
__global__ __launch_bounds__(256)
void plastic_bias(const v4f* __restrict__ grad,
                  const v4f* __restrict__ trace,
                  const v4f* __restrict__ mom,
                  const float* __restrict__ reward,
                  v4f* __restrict__ t_out,
                  v4f* __restrict__ m_out,
                  v4f* __restrict__ mod_out,
                  int n4)
{
    const int i = blockIdx.x * blockDim.x + threadIdx.x;
    if (i >= n4) return;
    const float r = reward[0] * MOD_STR;
    const v4f g = __builtin_nontemporal_load(grad  + i);
    const v4f t = __builtin_nontemporal_load(trace + i);
    const v4f m = __builtin_nontemporal_load(mom   + i);
    const v4f nt = TRACE_DECAY * t + 0.1f * g;
    const v4f nm = MOM_FACTOR * m + 0.05f * nt;
    const v4f md = r * nm;
    __builtin_nontemporal_store(nt, t_out + i);
    __builtin_nontemporal_store(nm, m_out + i);
    __builtin_nontemporal_store(md, mod_out + i);
}

extern "C" void kernel_launch(void* const* d_in, const int* in_sizes, int n_in,
                              void* d_out, int out_size, void* d_ws, size_t ws_size,
                              hipStream_t stream) {
    const float* grad_attn  = (const float*)d_in[0];
    const float* trace_attn = (const float*)d_in[1];
    const float* mom_attn   = (const float*)d_in[2];
    const float* U_attn     = (const float*)d_in[3];
    const float* V_attn     = (const float*)d_in[4];
    const float* grad_ff1   = (const float*)d_in[5];
    const float* trace_ff1  = (const float*)d_in[6];
    const float* mom_ff1    = (const float*)d_in[7];
    const float* U_ff1      = (const float*)d_in[8];
    const float* V_ff1      = (const float*)d_in[9];
    const float* grad_ff2   = (const float*)d_in[10];
    const float* trace_ff2  = (const float*)d_in[11];
    const float* mom_ff2    = (const float*)d_in[12];
    const float* U_ff2      = (const float*)d_in[13];
    const float* V_ff2      = (const float*)d_in[14];
    const float* grad_b     = (const float*)d_in[15];
    const float* trace_b    = (const float*)d_in[16];
    const float* mom_b      = (const float*)d_in[17];
    const float* reward     = (const float*)d_in[18];

    float* out = (float*)d_out;
    const size_t A = 2048ull * 2048ull;   // attn elems
    const size_t F = 8192ull * 2048ull;   // ff1/ff2 elems
    const size_t B = 8192ull;             // bias elems

    float* t_a   = out;
    float* m_a   = t_a + A;
    float* mod_a = m_a + A;
    float* t_1   = mod_a + A;
    float* m_1   = t_1 + F;
    float* mod_1 = m_1 + F;
    float* t_2   = mod_1 + F;
    float* m_2   = t_2 + F;
    float* mod_2 = m_2 + F;
    float* t_b   = mod_2 + F;
    float* m_b   = t_b + B;
    float* mod_b = m_b + B;

    const dim3 blk(256);

    // attn: M=2048, N=2048
    plastic_mat_wmma<2048><<<dim3(2048 / 128, 2048 / 16), blk, 0, stream>>>(
        grad_attn, trace_attn, mom_attn, U_attn, V_attn, reward,
        t_a, m_a, mod_a);

    // ff1: M=8192, N=2048
    plastic_mat_wmma<2048><<<dim3(2048 / 128, 8192 / 16), blk, 0, stream>>>(
        grad_ff1, trace_ff1, mom_ff1, U_ff1, V_ff1, reward,
        t_1, m_1, mod_1);

    // ff2: M=2048, N=8192
    plastic_mat_wmma<8192><<<dim3(8192 / 128, 2048 / 16), blk, 0, stream>>>(
        grad_ff2, trace_ff2, mom_ff2, U_ff2, V_ff2, reward,
        t_2, m_2, mod_2);

    // bias: 8192 elems -> 2048 float4
    plastic_bias<<<dim3((int)(B / 4 / 256)), blk, 0, stream>>>(
        (const v4f*)grad_b, (const v4f*)trace_b, (const v4f*)mom_b, reward,
        (v4f*)t_b, (v4f*)m_b, (v4f*)mod_b, (int)(B / 4));
}